// Model_28681791602755
// MI455X (gfx1250) — compile-verified
//
#include <hip/hip_runtime.h>
#include <hip/hip_bf16.h>

typedef __attribute__((ext_vector_type(2))) float v2f;
typedef __attribute__((ext_vector_type(8))) float v8f;
typedef int v4i __attribute__((vector_size(4 * sizeof(int))));
typedef __attribute__((address_space(1))) v4i* glob_v4i_ptr;   // HIP "__device__" AS
typedef __attribute__((address_space(3))) v4i* lds_v4i_ptr;    // HIP "__shared__" AS

// Linear-attention decode step:
//   out[b,h,e] = (q.k) * v[e] + exp(-slope[h]) * sum_d q[d]*kv_old[d,e]
// B=128, H=64, D=96, S=256.
// One wave32 per (b,h).  The 96x96 f32 kv tile (contiguous 36 KB) is streamed
// through LDS in 3 KB chunks with global_load_async_to_lds_b128 (double
// buffered, s_wait_asynccnt pipelining), and the gemv runs on
// V_WMMA_F32_16X16X4_F32 with ratio*q broadcast into all 16 A-rows.
#define BB 128
#define HH 64
#define DD 96
#define ROWS_PER_CHUNK 8
#define CHUNK_FLOATS (ROWS_PER_CHUNK * DD)   // 768 floats = 3 KB
#define NCHUNK (DD / ROWS_PER_CHUNK)         // 12 chunks

#if defined(__has_builtin)
#if __has_builtin(__builtin_amdgcn_global_load_async_to_lds_b128)
#define USE_ASYNC 1
#endif
#endif
#ifndef USE_ASYNC
#define USE_ASYNC 0
#endif

#if USE_ASYNC
#if __has_builtin(__builtin_amdgcn_s_wait_asynccnt)
#define WAIT_ASYNC(n) do { __builtin_amdgcn_s_wait_asynccnt(n); \
                           asm volatile("" ::: "memory"); } while (0)
#else
#define WAIT_ASYNC(n) asm volatile("s_wait_asynccnt %0" :: "i"(n) : "memory")
#endif
#else
#define WAIT_ASYNC(n) do {} while (0)
#endif

__global__ __launch_bounds__(256) void la_decode_wmma(
    const float* __restrict__ q,
    const float* __restrict__ k,
    const float* __restrict__ v,
    const float* __restrict__ kvc,
    const float* __restrict__ slope,
    const int*   __restrict__ slot,
    float* __restrict__ out)
{
    const int lane = threadIdx.x & 31;
    const int wave = threadIdx.x >> 5;
    const int p    = blockIdx.x * 8 + wave;      // pair index 0..8191
    const int b    = p >> 6;                     // p / H
    const int h    = p & 63;                     // p % H

    const int  sl    = slot[b];
    const bool valid = (sl >= 0);
    const int  ss    = valid ? sl : 0;
    const float ratio = __expf(-slope[h]);

    const float* qp  = q   + (size_t)p * DD;
    const float* kp  = k   + (size_t)p * DD;
    const float* vp  = v   + (size_t)p * DD;
    const float* kvp = kvc + ((size_t)ss * HH + h) * (size_t)(DD * DD);

    // per-wave double buffer: 8 waves * 2 * 3KB = 48 KB
    __shared__ float ldsbuf[8 * 2 * CHUNK_FLOATS];
    float* Lw = ldsbuf + wave * (2 * CHUNK_FLOATS);

    // ---- s = q . k  (wave32 shuffle reduction) ----
    float s = 0.0f;
    for (int i = lane; i < DD; i += 32)
        s += qp[i] * kp[i];
#pragma unroll
    for (int off = 16; off > 0; off >>= 1)
        s += __shfl_xor(s, off, 32);

    // ---- chunk stager: flat 3 KB copy, 6 x async b128 per wave ----
    auto stage = [&](int c, float* dst) {
        const float* src = kvp + (size_t)c * CHUNK_FLOATS;
#if USE_ASYNC
#pragma unroll
        for (int j = 0; j < 6; ++j) {
            const int off = (j * 32 + lane) * 4;
            __builtin_amdgcn_global_load_async_to_lds_b128(
                (glob_v4i_ptr)(const_cast<float*>(src + off)),
                (lds_v4i_ptr)(dst + off),
                /*offset=*/0, /*cpol=*/0);
        }
#else
#pragma unroll
        for (int j = 0; j < 6; ++j) {
            const int off = (j * 32 + lane) * 4;
            __builtin_prefetch(src + off + 2 * CHUNK_FLOATS, 0, 1);
            *reinterpret_cast<float4*>(dst + off) =
                *reinterpret_cast<const float4*>(src + off);
        }
#endif
    };

    // B-operand lane coords for the 4x16 f32 tile:
    //   lane half kh = lane>>4 covers K = 2*kh + vgprIdx;  N = lane & 15
    const int n  = lane & 15;
    const int kh = lane >> 4;

    v8f acc[6];
    const v8f zero = {0.f, 0.f, 0.f, 0.f, 0.f, 0.f, 0.f, 0.f};
#pragma unroll
    for (int eb = 0; eb < 6; ++eb) acc[eb] = zero;

    stage(0, Lw);
    for (int c = 0; c < NCHUNK; ++c) {
        if (c + 1 < NCHUNK) {
            stage(c + 1, Lw + ((c + 1) & 1) * CHUNK_FLOATS);
            WAIT_ASYNC(6);   // next chunk's 6 loads outstanding; chunk c done
        } else {
            WAIT_ASYNC(0);
        }
        const float* Lc = Lw + (c & 1) * CHUNK_FLOATS;
#pragma unroll
        for (int t = 0; t < 2; ++t) {
            const int d0 = c * ROWS_PER_CHUNK + t * 4;
            // A[m,k] = ratio * q[d0+k], identical for every row m.
            // 16x4 f32 A layout: lanes 0-15 -> (K0,K1); lanes 16-31 -> (K2,K3).
            const float4 qq = *reinterpret_cast<const float4*>(qp + d0);
            v2f A;
            A.x = (lane < 16 ? qq.x : qq.z) * ratio;
            A.y = (lane < 16 ? qq.y : qq.w) * ratio;

            const float* rowp = Lc + (t * 4 + 2 * kh) * DD + n;
#pragma unroll
            for (int eb = 0; eb < 6; ++eb) {
                v2f Bv;
                Bv.x = rowp[eb * 16];        // K = 2*kh + 0
                Bv.y = rowp[eb * 16 + DD];   // K = 2*kh + 1
                acc[eb] = __builtin_amdgcn_wmma_f32_16x16x4_f32(
                    /*neg_a=*/false, A, /*neg_b=*/false, Bv,
                    /*c_mod=*/(short)0, acc[eb],
                    /*reuse_a=*/false, /*reuse_b=*/false);
            }
        }
    }

    // ---- epilogue: D row M=0 lives in acc component 0, lanes 0..15 ----
    float* op = out + (size_t)p * DD;
    if (lane < 16) {
#pragma unroll
        for (int eb = 0; eb < 6; ++eb) {
            const int e = eb * 16 + lane;
            float r = valid ? (acc[eb][0] + s * vp[e]) : 0.0f;
            op[e] = r;
        }
    }
}

extern "C" void kernel_launch(void* const* d_in, const int* in_sizes, int n_in,
                              void* d_out, int out_size, void* d_ws, size_t ws_size,
                              hipStream_t stream) {
    const float* q     = (const float*)d_in[0];
    const float* k     = (const float*)d_in[1];
    const float* v     = (const float*)d_in[2];
    const float* kvc   = (const float*)d_in[3];
    const float* slope = (const float*)d_in[4];
    const int*   slot  = (const int*)  d_in[5];
    float*       out   = (float*)d_out;

    // 8192 (b,h) pairs, 1 wave each, 8 waves (256 threads) per block
    la_decode_wmma<<<dim3(1024), dim3(256), 0, stream>>>(q, k, v, kvc, slope, slot, out);
}